// NGCF_dgl_81398220194345
// MI455X (gfx1250) — compile-verified
//
#include <hip/hip_runtime.h>

typedef float v2f __attribute__((ext_vector_type(2)));
typedef float v8f __attribute__((ext_vector_type(8)));

#define DIMD 64
#define WPITCH 66  // LDS row pitch for weights (bank-conflict padding)

// ---------------- utility: zero fill ----------------
__global__ void zero_kernel(float* __restrict__ p, long n) {
  long i = (long)blockIdx.x * blockDim.x + threadIdx.x;
  long stride = (long)gridDim.x * blockDim.x;
  for (; i < n; i += stride) p[i] = 0.0f;
}

// ---------------- degree: deg[src]++ ----------------
__global__ void deg_kernel(const int* __restrict__ src, float* __restrict__ deg, int E) {
  int e = blockIdx.x * blockDim.x + threadIdx.x;
  if (e < E) atomicAdd(&deg[src[e]], 1.0f);
}

// ---------------- per-edge inv weight + inv_sum at dst ----------------
__global__ void inv_kernel(const int* __restrict__ src, const int* __restrict__ dst,
                           const float* __restrict__ deg, float* __restrict__ inv_e,
                           float* __restrict__ inv_sum, int E) {
  int e = blockIdx.x * blockDim.x + threadIdx.x;
  if (e >= E) return;
  int s = src[e], d = dst[e];
  float w = 1.0f / (sqrtf(deg[s]) * sqrtf(deg[d]));
  inv_e[e] = w;
  atomicAdd(&inv_sum[d], w);
}

// ---------------- SpMM: agg[dst] += inv[e] * emb[src]  (one wave per edge, 2 cols/lane)
__global__ void spmm_kernel(const int* __restrict__ src, const int* __restrict__ dst,
                            const float* __restrict__ inv_e, const float* __restrict__ emb,
                            float* __restrict__ agg, int E) {
  long t = (long)blockIdx.x * blockDim.x + threadIdx.x;
  long e = t >> 5;
  if (e >= E) return;
  int lane = (int)(t & 31);
  int s = src[e], d = dst[e];
  float w = inv_e[e];
  float2 v = *(const float2*)(emb + (long)s * DIMD + 2 * lane);
  float* ap = agg + (long)d * DIMD + 2 * lane;
  atomicAdd(ap, v.x * w);
  atomicAdd(ap + 1, v.y * w);
}

// ---------------- dense transform + leaky-relu + row L2 normalize ----------------
// out = (emb*(1+inv_sum)) @ Wg^T + (emb*agg) @ We^T  via V_WMMA_F32_16X16X4_F32
__global__ __launch_bounds__(256) void dense_kernel(
    const float* __restrict__ emb, const float* __restrict__ agg,
    const float* __restrict__ inv_sum,
    const float* __restrict__ Wg, const float* __restrict__ We,
    float* __restrict__ out_emb, int N) {
  __shared__ float sW[2][DIMD * WPITCH];
  for (int i = threadIdx.x; i < DIMD * DIMD; i += blockDim.x) {
    int r = i >> 6, c = i & 63;
    sW[0][r * WPITCH + c] = Wg[i];
    sW[1][r * WPITCH + c] = We[i];
  }
  __syncthreads();

  int wave = threadIdx.x >> 5;
  int lane = threadIdx.x & 31;
  int row0 = blockIdx.x * 128 + wave * 16;
  if (row0 + 15 >= N) {
    if (row0 >= N) return;  // (N is a multiple of 16 here; partial tiles skipped)
  }

  int hi = lane >> 4;   // which K-pair half of the wave
  int lr = lane & 15;   // M row (A) / N col (B, C/D)
  int arow = row0 + lr;
  float s1 = 1.0f + inv_sum[arow];

  const float* embr = emb + (long)arow * DIMD;
  const float* aggr = agg + (long)arow * DIMD;

  v8f c[4] = {};  // 4 output tiles of 16x16 along the 64 output columns

#pragma unroll
  for (int k4 = 0; k4 < DIMD; k4 += 4) {
    int ko = k4 + 2 * hi;
    float2 ev = *(const float2*)(embr + ko);
    float2 av = *(const float2*)(aggr + ko);
    v2f a1, a2;
    a1.x = ev.x * s1;    a1.y = ev.y * s1;     // (r1 + emb) operand
    a2.x = ev.x * av.x;  a2.y = ev.y * av.y;   // r2 operand
#pragma unroll
    for (int nt = 0; nt < 4; ++nt) {
      v2f bg = *(const v2f*)(&sW[0][(nt * 16 + lr) * WPITCH + ko]);  // B[k][n]=Wg[n][k]
      v2f be = *(const v2f*)(&sW[1][(nt * 16 + lr) * WPITCH + ko]);
      c[nt] = __builtin_amdgcn_wmma_f32_16x16x4_f32(false, a1, false, bg,
                                                    (short)0, c[nt], false, false);
      c[nt] = __builtin_amdgcn_wmma_f32_16x16x4_f32(false, a2, false, be,
                                                    (short)0, c[nt], false, false);
    }
  }

  // leaky relu (slope 0.2) + accumulate row sum-of-squares per accumulator reg
  float ssq[8];
#pragma unroll
  for (int v = 0; v < 8; ++v) ssq[v] = 0.0f;
#pragma unroll
  for (int nt = 0; nt < 4; ++nt) {
#pragma unroll
    for (int v = 0; v < 8; ++v) {
      float x = c[nt][v];
      x = (x > 0.0f) ? x : 0.2f * x;
      c[nt][v] = x;
      ssq[v] += x * x;
    }
  }
  // reduce across the 16 lanes of each half-wave (each half holds one row per VGPR)
#pragma unroll
  for (int v = 0; v < 8; ++v) {
    float s = ssq[v];
    s += __shfl_xor(s, 1, 32);
    s += __shfl_xor(s, 2, 32);
    s += __shfl_xor(s, 4, 32);
    s += __shfl_xor(s, 8, 32);
    ssq[v] = 1.0f / fmaxf(sqrtf(s), 1e-12f);
  }
  // D layout: lane -> N = lr (+16*nt), VGPR v -> M = v + 8*hi
#pragma unroll
  for (int nt = 0; nt < 4; ++nt) {
#pragma unroll
    for (int v = 0; v < 8; ++v) {
      out_emb[(long)(row0 + v + 8 * hi) * DIMD + nt * 16 + lr] = c[nt][v] * ssq[v];
    }
  }
}

// ---------------- gather one 64-col slice of all_emb into d_out ----------------
__global__ void gather_kernel(const float* __restrict__ embsrc,
                              const int* __restrict__ user, const int* __restrict__ pos,
                              const int* __restrict__ neg, const int* __restrict__ nup,
                              float* __restrict__ out, int col0, int ncol, int B) {
  long t = (long)blockIdx.x * blockDim.x + threadIdx.x;
  long total = (long)3 * B * DIMD;
  if (t >= total) return;
  int c = (int)(t & (DIMD - 1));
  long bp = t >> 6;
  int b = (int)(bp % B);
  int p = (int)(bp / B);
  int nu = nup[0];
  int row = (p == 0) ? user[b] : (nu + ((p == 1) ? pos[b] : neg[b]));
  out[((long)p * B + b) * ncol + col0 + c] = embsrc[(long)row * DIMD + c];
}

extern "C" void kernel_launch(void* const* d_in, const int* in_sizes, int n_in,
                              void* d_out, int out_size, void* d_ws, size_t ws_size,
                              hipStream_t stream) {
  const float* feature = (const float*)d_in[0];
  const float* W_gcn   = (const float*)d_in[1];
  const float* W_enh   = (const float*)d_in[2];
  const int*   esrc    = (const int*)d_in[3];
  const int*   edst    = (const int*)d_in[4];
  const int*   user    = (const int*)d_in[5];
  const int*   pos     = (const int*)d_in[6];
  const int*   neg     = (const int*)d_in[7];
  const int*   nup     = (const int*)d_in[8];

  int N = in_sizes[0] / DIMD;
  int L = in_sizes[1] / (DIMD * DIMD);
  int E = in_sizes[3];
  int B = in_sizes[5];
  int ncol = (L + 1) * DIMD;

  float* ws      = (float*)d_ws;
  float* deg     = ws;
  float* inv_sum = deg + N;
  float* inv_e   = inv_sum + N;
  float* embA    = inv_e + E;
  float* embB    = embA + (long)N * DIMD;
  float* agg     = embB + (long)N * DIMD;

  zero_kernel<<<2048, 256, 0, stream>>>(deg, 2L * N);  // deg + inv_sum contiguous
  deg_kernel<<<(E + 255) / 256, 256, 0, stream>>>(esrc, deg, E);
  inv_kernel<<<(E + 255) / 256, 256, 0, stream>>>(esrc, edst, deg, inv_e, inv_sum, E);

  long gth = 3L * B * DIMD;
  gather_kernel<<<(int)((gth + 255) / 256), 256, 0, stream>>>(
      feature, user, pos, neg, nup, (float*)d_out, 0, ncol, B);

  const float* cur = feature;
  float* nxt = embA;
  for (int l = 0; l < L; ++l) {
    zero_kernel<<<4096, 256, 0, stream>>>(agg, (long)N * DIMD);
    long st = (long)E * 32;
    spmm_kernel<<<(int)((st + 255) / 256), 256, 0, stream>>>(esrc, edst, inv_e, cur, agg, E);
    dense_kernel<<<(N + 127) / 128, 256, 0, stream>>>(
        cur, agg, inv_sum,
        W_gcn + (long)l * DIMD * DIMD, W_enh + (long)l * DIMD * DIMD, nxt, N);
    gather_kernel<<<(int)((gth + 255) / 256), 256, 0, stream>>>(
        nxt, user, pos, neg, nup, (float*)d_out, (l + 1) * DIMD, ncol, B);
    cur = nxt;
    nxt = (nxt == embA) ? embB : embA;
  }
}